// DNN_WPE_85177791414850
// MI455X (gfx1250) — compile-verified
//
#include <hip/hip_runtime.h>

// Problem constants (from reference)
#define NB 8
#define NF 257
#define NC 8
#define NT 800
#define KK 40          // taps * C
#define KP 48          // augmented width: 40 R cols + 8 P cols (= 3 x 16 tiles)
#define TT0 7          // delay + taps - 1
#define YS 824         // LDS row stride (824 % 64 = 56 -> channel rows hit distinct banks)
#define NBF (NB * NF)
#define NCHUNK 199     // ceil((NT - TT0) / 4) WMMA k-steps of 4 frames

typedef __attribute__((ext_vector_type(2))) float v2f;
typedef __attribute__((ext_vector_type(8))) float v8f;

#if defined(__HIP_DEVICE_COMPILE__)
#if !__has_builtin(__builtin_amdgcn_wmma_f32_16x16x4_f32)
#error "gfx1250 wmma f32 16x16x4 builtin not available"
#endif
#endif

__device__ __forceinline__ v8f wmma4(v2f a, v2f b, v8f c) {
  // D = A(16x4,f32) * B(4x16,f32) + C(16x16,f32)
  return __builtin_amdgcn_wmma_f32_16x16x4_f32(false, a, false, b, (short)0, c,
                                               false, false);
}

// Async global -> LDS 16-byte copy (per lane), tracked by ASYNCcnt.
// LDS aperture keeps the byte offset in addr[31:0], so truncating the generic
// shared pointer yields the VDST LDS address.
__device__ __forceinline__ void async_copy_b128(const float* gsrc, float* ldst) {
  unsigned int lds = (unsigned int)(unsigned long long)ldst;
  asm volatile("global_load_async_to_lds_b128 %0, %1, off"
               :
               : "v"(lds), "v"(gsrc)
               : "memory");
}
__device__ __forceinline__ void wait_async0() {
  asm volatile("s_wait_asynccnt 0x0" ::: "memory");
}

// ---------------------------------------------------------------------------
// Kernel 1: per (b,f) block of 96 threads (3 waves, wave32):
//   - async-DMA Y(mix) into LDS while computing 1/power weights (overlapped)
//   - accumulate augmented [R | P] (40x48 complex) with f32 WMMA
//   - Gauss-Jordan solve in LDS (reusing the Y region) -> G = R^{-1} P
// ---------------------------------------------------------------------------
__global__ void __launch_bounds__(96) wpe_rp_solve(
    const float* __restrict__ sep_r, const float* __restrict__ sep_i,
    const float* __restrict__ mix_r, const float* __restrict__ mix_i,
    float* __restrict__ Gbuf) {
  extern __shared__ float sm[];
  float* yr = sm;                 // NC * YS
  float* yi = yr + NC * YS;       // NC * YS
  float* w  = yi + NC * YS;       // 808 (zero-padded tail)
  // Reused after accumulation (Y region dead by then):
  float* Mr  = sm;                // KK * KP
  float* Mi  = sm + KK * KP;      // KK * KP
  float* piv = sm + 2 * KK * KP;  // 2 floats

  const int bf   = blockIdx.x;
  const int tid  = threadIdx.x;
  const int wave = tid >> 5;      // row-tile I in 0..2
  const int lane = tid & 31;
  const size_t base = (size_t)bf * NC * NT;

  // ---- kick off async Y staging (no VGPR round-trip, ASYNCcnt tracked) ----
  for (int i = tid; i < NC * (NT / 4); i += 96) {
    int c = i / (NT / 4), t4 = i - c * (NT / 4);
    async_copy_b128(mix_r + base + (size_t)c * NT + 4 * t4, yr + c * YS + 4 * t4);
    async_copy_b128(mix_i + base + (size_t)c * NT + 4 * t4, yi + c * YS + 4 * t4);
  }
  for (int i = tid; i < NC * (YS - NT); i += 96) {  // zero pad cols [NT, YS)
    int c = i / (YS - NT), t = NT + (i - c * (YS - NT));
    yr[c * YS + t] = 0.f;
    yi[c * YS + t] = 0.f;
  }
  // weights 1/max(mean_c |sep|^2, 1e-7) -- overlaps with the async DMA above;
  // zero past T (kills the remainder chunk contribution)
  for (int t = tid; t < 808; t += 96) {
    float wv = 0.f;
    if (t < NT) {
      float p = 0.f;
      for (int c = 0; c < NC; ++c) {
        float a = sep_r[base + c * NT + t];
        float b = sep_i[base + c * NT + t];
        p += a * a + b * b;
      }
      wv = 1.f / fmaxf(p * (1.f / NC), 1e-7f);
    }
    w[t] = wv;
  }
  wait_async0();
  __syncthreads();

  // ---- WMMA accumulation of [R | P] ----
  // A fragment lane mapping (16x4 f32): m = lane&15, K pair = 2*(lane>=16)
  const int m    = lane & 15;
  const int kk0  = (lane >> 4) << 1;
  const int arow = wave * 16 + m;          // row a of Ytw
  const bool aval = arow < KK;
  const int ak = aval ? (arow >> 3) : 0;   // tap index
  const int ac = aval ? (arow & 7) : 0;    // channel
  const int offA = ac * YS - 3 - ak;       // Yt[a,t] = Y[c][t-3-k]
  const float mA = aval ? 1.f : 0.f;

  int offB[3];                             // B cols: augmented col j = 16J + n
#pragma unroll
  for (int J = 0; J < 3; ++J) {
    int j = J * 16 + m;
    offB[J] = (j < KK) ? ((j & 7) * YS - 3 - (j >> 3))  // conj(Ytv) col
                       : ((j - KK) * YS);               // conj(Yv) col (P)
  }

  const v8f vz = {0.f, 0.f, 0.f, 0.f, 0.f, 0.f, 0.f, 0.f};
  v8f acc[3][4];
#pragma unroll
  for (int J = 0; J < 3; ++J)
#pragma unroll
    for (int q = 0; q < 4; ++q) acc[J][q] = vz;

  for (int ch = 0; ch < NCHUNK; ++ch) {
    const int tvx = TT0 + 4 * ch + kk0;
    const float wx = w[tvx] * mA;
    const float wy = w[tvx + 1] * mA;
    v2f war, wai;
    war.x = yr[offA + tvx] * wx;
    war.y = yr[offA + tvx + 1] * wy;
    wai.x = yi[offA + tvx] * wx;
    wai.y = yi[offA + tvx + 1] * wy;
#pragma unroll
    for (int J = 0; J < 3; ++J) {
      v2f br, bi;
      br.x = yr[offB[J] + tvx];
      br.y = yr[offB[J] + tvx + 1];
      bi.x = yi[offB[J] + tvx];
      bi.y = yi[offB[J] + tvx + 1];
      acc[J][0] = wmma4(war, br, acc[J][0]);  // sum w*Ar*Ar'
      acc[J][1] = wmma4(wai, bi, acc[J][1]);  // sum w*Ai*Ai'
      acc[J][2] = wmma4(wai, br, acc[J][2]);  // sum w*Ai*Ar'
      acc[J][3] = wmma4(war, bi, acc[J][3]);  // sum w*Ar*Ai'
    }
  }
  __syncthreads();  // Y region dead; reuse LDS for augmented matrix

  // C/D layout: vgpr r, lane l -> row = r + 8*(l>=16), col = l&15
  const int rhi = (lane >> 4) * 8;
#pragma unroll
  for (int J = 0; J < 3; ++J) {
    v8f Rr = acc[J][0] + acc[J][1];  // real part
    v8f Ri = acc[J][2] - acc[J][3];  // imag part
#pragma unroll
    for (int r = 0; r < 8; ++r) {
      int a = wave * 16 + rhi + r;
      int j = J * 16 + m;
      if (a < KK) {
        float rr = Rr[r];
        if (j == a) rr += 1e-10f;  // EPS_REG * I on R's diagonal
        Mr[a * KP + j] = rr;
        Mi[a * KP + j] = Ri[r];
      }
    }
  }
  __syncthreads();

  // ---- complex Gauss-Jordan on [R | P] (R Hermitian + eps: no pivoting) ----
  for (int p = 0; p < KK; ++p) {
    if (tid == 0) {
      float dr = Mr[p * KP + p], di = Mi[p * KP + p];
      float den = 1.f / (dr * dr + di * di + 1e-30f);
      piv[0] = dr * den;
      piv[1] = -di * den;
    }
    __syncthreads();
    if (tid < KP) {  // scale pivot row
      float sr = piv[0], si = piv[1];
      float xr = Mr[p * KP + tid], xi = Mi[p * KP + tid];
      Mr[p * KP + tid] = xr * sr - xi * si;
      Mi[p * KP + tid] = xr * si + xi * sr;
    }
    __syncthreads();
    if (tid < KK && tid != p) {  // eliminate: each thread owns a row
      int r = tid;
      float fr = Mr[r * KP + p], fi = Mi[r * KP + p];
      for (int j = 0; j < KP; ++j) {
        float pr = Mr[p * KP + j], pi2 = Mi[p * KP + j];
        Mr[r * KP + j] -= fr * pr - fi * pi2;
        Mi[r * KP + j] -= fr * pi2 + fi * pr;
      }
    }
    __syncthreads();
  }

  // cols 40..47 now hold G = R^{-1} P ; store interleaved (re,im)
  for (int i = tid; i < KK * NC; i += 96) {
    int a = i >> 3, e = i & 7;
    float* g = Gbuf + ((size_t)bf * KK * NC + i) * 2;
    g[0] = Mr[a * KP + KK + e];
    g[1] = Mi[a * KP + KK + e];
  }
}

// ---------------------------------------------------------------------------
// Kernel 2: per (b,f) block of 128 threads (4 waves):
//   X[e,t] = Y[e,t] - sum_a conj(G[a,e]) * Yt[a,t], masked by ilens, via WMMA
//   (A = conj(G)^T 8x40 padded to 16, B = Yt k-slices, 50 t-tiles of 16)
// ---------------------------------------------------------------------------
__global__ void __launch_bounds__(128) wpe_apply(
    const float* __restrict__ mix_r, const float* __restrict__ mix_i,
    const float* __restrict__ Gbuf, const int* __restrict__ ilens,
    float* __restrict__ out) {
  extern __shared__ float sm[];
  float* yr = sm;             // NC * YS, data at cols [8, 808), zero pads outside
  float* yi = yr + NC * YS;
  float* gr = yi + NC * YS;   // KK * NC
  float* gi = gr + KK * NC;

  const int bf = blockIdx.x;
  const int tid = threadIdx.x;
  const int wave = tid >> 5, lane = tid & 31;
  const size_t base = (size_t)bf * NC * NT;
  const int ilen = ilens[bf / NF];

  // async Y staging overlapped with the (regular) G fetch below
  for (int i = tid; i < NC * (NT / 4); i += 128) {
    int c = i / (NT / 4), t4 = i - c * (NT / 4);
    async_copy_b128(mix_r + base + (size_t)c * NT + 4 * t4,
                    yr + c * YS + 8 + 4 * t4);
    async_copy_b128(mix_i + base + (size_t)c * NT + 4 * t4,
                    yi + c * YS + 8 + 4 * t4);
  }
  for (int i = tid; i < NC * (YS - NT); i += 128) {  // zero cols [0,8) + [808,YS)
    int c = i / (YS - NT), t = i - c * (YS - NT);
    int col = (t < 8) ? t : (NT + t);
    yr[c * YS + col] = 0.f;
    yi[c * YS + col] = 0.f;
  }
  for (int i = tid; i < KK * NC; i += 128) {
    gr[i] = Gbuf[((size_t)bf * KK * NC + i) * 2 + 0];
    gi[i] = Gbuf[((size_t)bf * KK * NC + i) * 2 + 1];
  }
  wait_async0();
  __syncthreads();

  // A fragments from G (hoisted out of the t-tile loop): rows m = e (8 valid)
  const int me = lane & 15;
  const int kk0 = (lane >> 4) << 1;
  const float mA = (me < NC) ? 1.f : 0.f;
  const int ge = me & 7;
  v2f agr[10], agi[10], agin[10];
  int o0[10], o1[10];
#pragma unroll
  for (int kt = 0; kt < 10; ++kt) {
    int a0 = 4 * kt + kk0, a1 = a0 + 1;
    agr[kt].x = gr[a0 * NC + ge] * mA;
    agr[kt].y = gr[a1 * NC + ge] * mA;
    agi[kt].x = gi[a0 * NC + ge] * mA;
    agi[kt].y = gi[a1 * NC + ge] * mA;
    agin[kt].x = -agi[kt].x;
    agin[kt].y = -agi[kt].y;
    // Yt[a,t] lives at LDS col t + 5 - tap(a) in channel row a&7
    o0[kt] = (a0 & 7) * YS + 5 - (a0 >> 3);
    o1[kt] = (a1 & 7) * YS + 5 - (a1 >> 3);
  }

  const v8f vz = {0.f, 0.f, 0.f, 0.f, 0.f, 0.f, 0.f, 0.f};
  const int nn = lane & 15;
  for (int tt = wave; tt < NT / 16; tt += 4) {
    const int t = tt * 16 + nn;
    v8f accR = vz, accI = vz;
#pragma unroll
    for (int kt = 0; kt < 10; ++kt) {
      v2f byr, byi;
      byr.x = yr[o0[kt] + t];
      byr.y = yr[o1[kt] + t];
      byi.x = yi[o0[kt] + t];
      byi.y = yi[o1[kt] + t];
      accR = wmma4(agr[kt], byr, accR);   // Gr*Ytr
      accR = wmma4(agi[kt], byi, accR);   // + Gi*Yti
      accI = wmma4(agr[kt], byi, accI);   // Gr*Yti
      accI = wmma4(agin[kt], byr, accI);  // - Gi*Ytr
    }
    if (lane < 16) {  // rows 0..7 (= channels) live in lanes 0..15
#pragma unroll
      for (int r = 0; r < 8; ++r) {
        float xr = yr[r * YS + 8 + t] - accR[r];
        float xi = yi[r * YS + 8 + t] - accI[r];
        bool valid = t < ilen;
        size_t o = (((size_t)bf * NC + r) * NT + t) * 2;
        out[o] = valid ? xr : 0.f;
        out[o + 1] = valid ? xi : 0.f;
      }
    }
  }
}

// ---------------------------------------------------------------------------
extern "C" void kernel_launch(void* const* d_in, const int* in_sizes, int n_in,
                              void* d_out, int out_size, void* d_ws,
                              size_t ws_size, hipStream_t stream) {
  const float* sep_r = (const float*)d_in[0];
  const float* sep_i = (const float*)d_in[1];
  const float* mix_r = (const float*)d_in[2];
  const float* mix_i = (const float*)d_in[3];
  const int* ilens = (const int*)d_in[4];
  float* Gbuf = (float*)d_ws;  // NBF * 40 * 8 * 2 floats ~= 5.3 MB

  size_t smem1 = (size_t)(2 * NC * YS + 808 + 8) * sizeof(float);    // 56 KB
  size_t smem2 = (size_t)(2 * NC * YS + 2 * KK * NC) * sizeof(float);// 55 KB
  wpe_rp_solve<<<NBF, 96, smem1, stream>>>(sep_r, sep_i, mix_r, mix_i, Gbuf);
  wpe_apply<<<NBF, 128, smem2, stream>>>(mix_r, mix_i, Gbuf, ilens,
                                         (float*)d_out);
}